// PatchSampleF_QSAttn_11544872092295
// MI455X (gfx1250) — compile-verified
//
#include <hip/hip_runtime.h>

typedef __attribute__((ext_vector_type(2))) float v2f;
typedef __attribute__((ext_vector_type(8))) float v8f;

#define P_SEL 64
#define BQ 2
#define CQ 256
#define HQ 64
#define WQ 64
#define LQ (HQ * WQ)

// ---------------- sample_early: gather fixed patch ids + L2 normalize ----------------
__global__ void sample_kernel(const float* __restrict__ feat, const int* __restrict__ pid,
                              int C, int HW, float* __restrict__ out) {
  int row = blockIdx.x;          // [0, 2*64)
  int b = row >> 6;
  int j = row & 63;
  int l = pid[j];
  const float* src = feat + (size_t)b * C * HW + l;   // fr[b, l, c] = feat[b, c, l]
  float* dst = out + (size_t)row * C;
  __shared__ float red[256];
  float ss = 0.f;
  for (int c = (int)threadIdx.x; c < C; c += blockDim.x) {
    float v = src[(size_t)c * HW];
    ss += v * v;
  }
  red[threadIdx.x] = ss;
  __syncthreads();
  for (int s = blockDim.x >> 1; s > 0; s >>= 1) {
    if ((int)threadIdx.x < s) red[threadIdx.x] += red[threadIdx.x + s];
    __syncthreads();
  }
  float inv = 1.0f / fmaxf(sqrtf(red[0]), 1e-7f);
  for (int c = (int)threadIdx.x; c < C; c += blockDim.x)
    dst[c] = src[(size_t)c * HW] * inv;
}

// ---------------- local-window dots (with faithful permute/reshape scramble) + entropy ----------------
// feat_k_local[b,l,k,c] = feat_local[b, k*C + c, l]; feat_local flat index = c2*49 + k2
__global__ void entropy_kernel(const float* __restrict__ feat, float* __restrict__ ent) {
  int tid = blockIdx.x * blockDim.x + threadIdx.x;
  if (tid >= BQ * LQ) return;
  int b = tid / LQ;
  int l = tid - b * LQ;
  int h = l / WQ, w = l - h * WQ;
  const float* fb = feat + (size_t)b * CQ * LQ;
  float dots[49];
  for (int k = 0; k < 49; ++k) {
    int flat = k * CQ;
    int c2 = flat / 49;
    int k2 = flat - c2 * 49;
    float acc = 0.f;
    for (int c = 0; c < CQ; ++c) {
      int i2 = k2 / 7;
      int j2 = k2 - i2 * 7;
      int hn = h + i2 - 3;
      int wn = w + j2 - 3;
      float nb = 0.f;
      if (hn >= 0 && hn < HQ && wn >= 0 && wn < WQ)
        nb = fb[(size_t)c2 * LQ + hn * WQ + wn];
      acc += nb * fb[(size_t)c * LQ + l];
      if (++k2 == 49) { k2 = 0; ++c2; }
    }
    dots[k] = acc;
  }
  float m = dots[0];
  for (int k = 1; k < 49; ++k) m = fmaxf(m, dots[k]);
  float s = 0.f, sd = 0.f;
  for (int k = 0; k < 49; ++k) {
    float e = expf(dots[k] - m);
    s += e;
    sd += dots[k] * e;
  }
  float lse = m + logf(s);
  ent[tid] = lse - sd / s;      // sum_k p_k * (lse - d_k)
}

// ---------------- lowest-64-entropy selection per batch (stable: smallest index on ties) ----------------
__global__ void select_kernel(const float* __restrict__ ent, int* __restrict__ idx) {
  int b = blockIdx.x;
  const float* e = ent + (size_t)b * LQ;
  __shared__ float sval[LQ];      // 16 KB
  __shared__ float bv[256];
  __shared__ int   bi[256];
  for (int i = (int)threadIdx.x; i < LQ; i += blockDim.x) sval[i] = e[i];
  __syncthreads();
  for (int p = 0; p < P_SEL; ++p) {
    float best = __builtin_inff();
    int besti = LQ;
    for (int i = (int)threadIdx.x; i < LQ; i += blockDim.x) {
      float v = sval[i];
      if (v < best || (v == best && i < besti)) { best = v; besti = i; }
    }
    bv[threadIdx.x] = best;
    bi[threadIdx.x] = besti;
    __syncthreads();
    if (threadIdx.x == 0) {
      float gb = bv[0]; int gi = bi[0];
      for (int t = 1; t < (int)blockDim.x; ++t)
        if (bv[t] < gb || (bv[t] == gb && bi[t] < gi)) { gb = bv[t]; gi = bi[t]; }
      idx[b * P_SEL + p] = gi;
      sval[gi] = __builtin_inff();
    }
    __syncthreads();
  }
}

// ---------------- GEMM1: dots[b,p,l] = sum_c q[p,c] * fr[l,c]  (WMMA f32 16x16x4) ----------------
__global__ void gemm1_wmma(const float* __restrict__ feat, const int* __restrict__ idx,
                           float* __restrict__ dots) {
  const int ntl = LQ >> 4;               // 256 l-tiles
  int blk = blockIdx.x;
  int lt = blk % ntl;
  int tmp = blk / ntl;
  int pt = tmp & 3;                      // 4 p-tiles of 16
  int b  = tmp >> 2;
  int lane = threadIdx.x;                // wave32
  int g = lane >> 4;
  int m = lane & 15;
  const float* fb = feat + (size_t)b * CQ * LQ;
  int qp = idx[b * P_SEL + pt * 16 + m];         // A row: gathered patch id
  int ln = lt * 16 + m;                          // B col: position id
  v8f acc = {};
  for (int k = 0; k < CQ; k += 4) {
    int k0 = k + 2 * g;
    v2f a, bm;
    a.x  = fb[(size_t)(k0)     * LQ + qp];
    a.y  = fb[(size_t)(k0 + 1) * LQ + qp];
    bm.x = fb[(size_t)(k0)     * LQ + ln];
    bm.y = fb[(size_t)(k0 + 1) * LQ + ln];
    acc = __builtin_amdgcn_wmma_f32_16x16x4_f32(false, a, false, bm, (short)0, acc, false, false);
  }
  int col = lt * 16 + m;
#pragma unroll
  for (int r = 0; r < 8; ++r) {
    int prow = pt * 16 + r + 8 * g;
    dots[((size_t)b * P_SEL + prow) * LQ + col] = acc[r];
  }
}

// ---------------- row softmax over L=4096, in place ----------------
__global__ void softmax_kernel(float* __restrict__ dots) {
  int row = blockIdx.x;                  // B*64 rows
  float* d = dots + (size_t)row * LQ;
  __shared__ float red[256];
  float m = -__builtin_inff();
  for (int i = (int)threadIdx.x; i < LQ; i += blockDim.x) m = fmaxf(m, d[i]);
  red[threadIdx.x] = m;
  __syncthreads();
  for (int s = blockDim.x >> 1; s > 0; s >>= 1) {
    if ((int)threadIdx.x < s) red[threadIdx.x] = fmaxf(red[threadIdx.x], red[threadIdx.x + s]);
    __syncthreads();
  }
  m = red[0];
  __syncthreads();
  float sl = 0.f;
  for (int i = (int)threadIdx.x; i < LQ; i += blockDim.x) {
    float e = expf(d[i] - m);
    d[i] = e;
    sl += e;
  }
  red[threadIdx.x] = sl;
  __syncthreads();
  for (int s = blockDim.x >> 1; s > 0; s >>= 1) {
    if ((int)threadIdx.x < s) red[threadIdx.x] += red[threadIdx.x + s];
    __syncthreads();
  }
  float inv = 1.0f / red[0];
  for (int i = (int)threadIdx.x; i < LQ; i += blockDim.x) d[i] *= inv;
}

// ---------------- GEMM2: out[b,p,c] = sum_l attn[p,l] * fr[l,c]  (WMMA f32 16x16x4) ----------------
__global__ void gemm2_wmma(const float* __restrict__ feat, const float* __restrict__ attn,
                           float* __restrict__ out3) {
  const int nct = CQ >> 4;               // 16 c-tiles
  int blk = blockIdx.x;
  int ct = blk % nct;
  int tmp = blk / nct;
  int pt = tmp & 3;
  int b  = tmp >> 2;
  int lane = threadIdx.x;
  int g = lane >> 4;
  int m = lane & 15;
  const float* fb = feat + (size_t)b * CQ * LQ;
  const float* arow = attn + ((size_t)b * P_SEL + pt * 16 + m) * LQ;  // A row (contiguous in k)
  int cn = ct * 16 + m;
  const float* bcol = fb + (size_t)cn * LQ;                            // B[k][n] = fr[k, cn]
  v8f acc = {};
  for (int k = 0; k < LQ; k += 4) {
    int k0 = k + 2 * g;
    v2f a, bm;
    a.x  = arow[k0];
    a.y  = arow[k0 + 1];
    bm.x = bcol[k0];
    bm.y = bcol[k0 + 1];
    acc = __builtin_amdgcn_wmma_f32_16x16x4_f32(false, a, false, bm, (short)0, acc, false, false);
  }
  int col = ct * 16 + m;
#pragma unroll
  for (int r = 0; r < 8; ++r) {
    int prow = b * P_SEL + pt * 16 + r + 8 * g;
    out3[(size_t)prow * CQ + col] = acc[r];
  }
}

// ---------------- final L2 normalize of x3 rows ----------------
__global__ void norm_rows(const float* __restrict__ src, float* __restrict__ dst, int C) {
  int row = blockIdx.x;
  const float* s = src + (size_t)row * C;
  float* d = dst + (size_t)row * C;
  __shared__ float red[256];
  float ss = 0.f;
  for (int c = (int)threadIdx.x; c < C; c += blockDim.x) { float v = s[c]; ss += v * v; }
  red[threadIdx.x] = ss;
  __syncthreads();
  for (int st = blockDim.x >> 1; st > 0; st >>= 1) {
    if ((int)threadIdx.x < st) red[threadIdx.x] += red[threadIdx.x + st];
    __syncthreads();
  }
  float inv = 1.0f / fmaxf(sqrtf(red[0]), 1e-7f);
  for (int c = (int)threadIdx.x; c < C; c += blockDim.x) d[c] = s[c] * inv;
}

extern "C" void kernel_launch(void* const* d_in, const int* in_sizes, int n_in,
                              void* d_out, int out_size, void* d_ws, size_t ws_size,
                              hipStream_t stream) {
  const float* feat0 = (const float*)d_in[0];   // [2,64,256,256]
  const float* feat1 = (const float*)d_in[1];   // [2,128,128,128]
  const float* feat2 = (const float*)d_in[2];   // [2,256,64,64]
  const float* feat3 = (const float*)d_in[3];   // [2,256,64,64]
  const int* pid0 = (const int*)d_in[4];
  const int* pid1 = (const int*)d_in[5];
  const int* pid2 = (const int*)d_in[6];
  float* out = (float*)d_out;

  float* ws   = (float*)d_ws;
  float* ent  = ws;                        // 2*4096 floats
  int*   idx  = (int*)(ws + 8192);         // 2*64 ints
  float* dots = ws + 8320;                 // 2*64*4096 floats (attn in place)
  float* o3   = ws + 8320 + 2 * P_SEL * LQ; // 2*64*256 floats (pre-norm x3)

  // outputs: x0 [128,64] | x1 [128,128] | x2 [128,256] | x3 [128,256]
  float* out0 = out;
  float* out1 = out0 + 128 * 64;
  float* out2 = out1 + 128 * 128;
  float* out3 = out2 + 128 * 256;

  sample_kernel<<<128, 256, 0, stream>>>(feat0, pid0, 64, 256 * 256, out0);
  sample_kernel<<<128, 256, 0, stream>>>(feat1, pid1, 128, 128 * 128, out1);
  sample_kernel<<<128, 256, 0, stream>>>(feat2, pid2, 256, 64 * 64, out2);

  entropy_kernel<<<(BQ * LQ + 255) / 256, 256, 0, stream>>>(feat3, ent);
  select_kernel<<<BQ, 256, 0, stream>>>(ent, idx);

  gemm1_wmma<<<BQ * 4 * (LQ / 16), 32, 0, stream>>>(feat3, idx, dots);
  softmax_kernel<<<BQ * P_SEL, 256, 0, stream>>>(dots);
  gemm2_wmma<<<BQ * 4 * (CQ / 16), 32, 0, stream>>>(feat3, dots, o3);
  norm_rows<<<BQ * P_SEL, 256, 0, stream>>>(o3, out3, CQ);
}